// PointNetSetAbstraction_47029891891547
// MI455X (gfx1250) — compile-verified
//
#include <hip/hip_runtime.h>
#include <math.h>

#define B_   8
#define N_   4096
#define D_   64
#define S_   1024
#define K_   32
#define R2_  0.04f
#define EPS_ 1e-5f

typedef __attribute__((ext_vector_type(2))) float v2f;
typedef __attribute__((ext_vector_type(8))) float v8f;

// CDNA5 fp32 matrix op: D(16x16) = A(16x4) * B(4x16) + C
__device__ __forceinline__ v8f wmma4(v2f a, v2f b, v8f c) {
  return __builtin_amdgcn_wmma_f32_16x16x4_f32(
      /*neg_a=*/false, a, /*neg_b=*/false, b,
      /*c_mod=*/(short)0, c, /*reuse_a=*/false, /*reuse_b=*/false);
}

// ---------------------------------------------------------------------------
// Kernel 1: farthest point sampling. One workgroup per batch; points in LDS,
// per-thread register dist, LDS tree argmax with first-index tie-break.
// Writes new_xyz (B,3,S) directly into d_out.
// ---------------------------------------------------------------------------
__global__ void __launch_bounds__(256) fps_kernel(const float* __restrict__ xyz,
                                                  float* __restrict__ newxyz) {
  __shared__ float px[N_], py[N_], pz[N_];
  __shared__ float sval[256];
  __shared__ int   sidx[256];
  __shared__ int   sfps[S_];
  const int b   = blockIdx.x;
  const int tid = threadIdx.x;
  const float* X = xyz + (size_t)b * 3 * N_;

  for (int p = tid; p < N_; p += 256) {
    px[p] = X[p];
    py[p] = X[N_ + p];
    pz[p] = X[2 * N_ + p];
  }
  if (tid == 0) sfps[0] = 0;
  __syncthreads();

  float dist[16];
#pragma unroll
  for (int j = 0; j < 16; ++j) dist[j] = 1e10f;

  int last = 0;
  for (int it = 1; it < S_; ++it) {
    const float lx = px[last], ly = py[last], lz = pz[last];
    float bv = -1.0f;
    int   bp = 0;
#pragma unroll
    for (int j = 0; j < 16; ++j) {
      const int p = tid + j * 256;           // ascending within thread
      const float dx = px[p] - lx, dy = py[p] - ly, dz = pz[p] - lz;
      const float d  = dx * dx + dy * dy + dz * dz;
      dist[j] = fminf(dist[j], d);
      if (dist[j] > bv) { bv = dist[j]; bp = p; }  // strict > keeps lowest idx
    }
    sval[tid] = bv;
    sidx[tid] = bp;
    __syncthreads();
    for (int off = 128; off > 0; off >>= 1) {
      if (tid < off) {
        const float v2 = sval[tid + off];
        const int   i2 = sidx[tid + off];
        if (v2 > sval[tid] || (v2 == sval[tid] && i2 < sidx[tid])) {
          sval[tid] = v2;
          sidx[tid] = i2;
        }
      }
      __syncthreads();
    }
    last = sidx[0];
    if (tid == 0) sfps[it] = last;
    __syncthreads();
  }

  for (int s = tid; s < S_; s += 256) {
    const int i = sfps[s];
    newxyz[(b * 3 + 0) * S_ + s] = px[i];
    newxyz[(b * 3 + 1) * S_ + s] = py[i];
    newxyz[(b * 3 + 2) * S_ + s] = pz[i];
  }
}

// ---------------------------------------------------------------------------
// Kernel 2: ball query. One wave32 per (b,s). Collect the first K_ point
// indices (ascending) with d2<=R2 via ballot+popcount prefix; pad with first.
// ---------------------------------------------------------------------------
__global__ void __launch_bounds__(32) ballquery_kernel(const float* __restrict__ xyz,
                                                       const float* __restrict__ newxyz,
                                                       int* __restrict__ idx) {
  const int bid  = blockIdx.x;
  const int b    = bid >> 10;
  const int s    = bid & (S_ - 1);
  const int lane = threadIdx.x;
  const float* X = xyz + (size_t)b * 3 * N_;
  const float qx = newxyz[(b * 3 + 0) * S_ + s];
  const float qy = newxyz[(b * 3 + 1) * S_ + s];
  const float qz = newxyz[(b * 3 + 2) * S_ + s];
  int* out = idx + (size_t)bid * K_;

  int count = 0;
  int first = -1;
  for (int base = 0; base < N_ && count < K_; base += 32) {
    const int p = base + lane;
    const float dx = X[p] - qx, dy = X[N_ + p] - qy, dz = X[2 * N_ + p] - qz;
    const bool pred = (dx * dx + dy * dy + dz * dz) <= R2_;
    const unsigned m = (unsigned)__ballot(pred);
    if (m) {
      if (first < 0) first = base + __ffs(m) - 1;
      const int pos = count + __popc(m & ((1u << lane) - 1u));
      if (pred && pos < K_) out[pos] = p;
      count += __popc(m);
    }
  }
  // query point itself is always within radius => first >= 0, count >= 1
  if (count < K_) {
    const int pos = count + lane;
    if (pos < K_) out[pos] = first;
  }
}

// ---------------------------------------------------------------------------
// Kernel 3: fused gather + 3x (1x1 conv + BN + ReLU) + max over K_ samples.
// One wave32 per (b,s): feature tile 68x32 in LDS, chained fp32 WMMA GEMMs.
// WMMAs run on RAW weights; BN is applied per output row in the epilogue:
//   y = scr * acc + bias, scr = g*rsqrt(rv+eps), bias = scr*(b-rm)+bt
// Fragment layouts per ISA 7.12.2 (16x16x4 f32):
//   A: lane l row = l&15, k = kb + 2*(l>>4) + {0,1}
//   B: lane l col = l&15, k = kb + 2*(l>>4) + {0,1}
//   D: vgpr v -> row v + 8*(l>>4), col = l&15
// ---------------------------------------------------------------------------
__global__ void __launch_bounds__(32) fused_mlp_kernel(
    const float* __restrict__ xyz, const float* __restrict__ points,
    const int* __restrict__ idx, const float* __restrict__ newxyz,
    const float* __restrict__ w0, const float* __restrict__ b0,
    const float* __restrict__ g0, const float* __restrict__ bt0,
    const float* __restrict__ rm0, const float* __restrict__ rv0,
    const float* __restrict__ w1, const float* __restrict__ b1,
    const float* __restrict__ g1, const float* __restrict__ bt1,
    const float* __restrict__ rm1, const float* __restrict__ rv1,
    const float* __restrict__ w2, const float* __restrict__ b2,
    const float* __restrict__ g2, const float* __restrict__ bt2,
    const float* __restrict__ rm2, const float* __restrict__ rv2,
    float* __restrict__ outF) {
  __shared__ float actA[68][K_];   // layer0 input (padded K=68) / layer2 input
  __shared__ float actB[64][K_];   // layer1 input
  const int bid  = blockIdx.x;
  const int b    = bid >> 10;
  const int s    = bid & (S_ - 1);
  const int lane = threadIdx.x;

  // ---- gather grouped features: ch 0..2 = xyz - new_xyz, ch 3..66 = points
  {
    const int pid  = idx[(size_t)bid * K_ + lane];
    const float* X = xyz + (size_t)b * 3 * N_;
    actA[0][lane] = X[pid]          - newxyz[(b * 3 + 0) * S_ + s];
    actA[1][lane] = X[N_ + pid]     - newxyz[(b * 3 + 1) * S_ + s];
    actA[2][lane] = X[2 * N_ + pid] - newxyz[(b * 3 + 2) * S_ + s];
    const float* P = points + (size_t)b * D_ * N_;
    for (int c = 0; c < D_; ++c)
      actA[3 + c][lane] = P[(size_t)c * N_ + pid];
    actA[67][lane] = 0.0f;           // K padding row
  }
  __syncthreads();

  const int half = lane >> 4;        // 0 or 1
  const int l15  = lane & 15;

  // ---- layer 0: 67(->68) -> 64, read actA, write actB
  for (int mt = 0; mt < 4; ++mt) {
    const int m = mt * 16 + l15;
    v8f c0 = {}; v8f c1 = {};
#pragma unroll
    for (int ks = 0; ks < 17; ++ks) {
      const int k0 = ks * 4 + 2 * half;
      v2f a, f0, f1;
      a.x  = (k0     < 67) ? w0[m * 67 + k0]     : 0.0f;
      a.y  = (k0 + 1 < 67) ? w0[m * 67 + k0 + 1] : 0.0f;
      f0.x = actA[k0][l15];        f0.y = actA[k0 + 1][l15];
      f1.x = actA[k0][16 + l15];   f1.y = actA[k0 + 1][16 + l15];
      c0 = wmma4(a, f0, c0);
      c1 = wmma4(a, f1, c1);
    }
#pragma unroll
    for (int v = 0; v < 8; ++v) {
      const int   r    = mt * 16 + v + 8 * half;
      const float scr  = g0[r] * rsqrtf(rv0[r] + EPS_);
      const float bias = scr * (b0[r] - rm0[r]) + bt0[r];
      actB[r][l15]      = fmaxf(fmaf(scr, c0[v], bias), 0.0f);
      actB[r][16 + l15] = fmaxf(fmaf(scr, c1[v], bias), 0.0f);
    }
  }
  __syncthreads();

  // ---- layer 1: 64 -> 64, read actB, write actA rows 0..63
  for (int mt = 0; mt < 4; ++mt) {
    const int m = mt * 16 + l15;
    v8f c0 = {}; v8f c1 = {};
#pragma unroll
    for (int ks = 0; ks < 16; ++ks) {
      const int k0 = ks * 4 + 2 * half;
      v2f a, f0, f1;
      a.x  = w1[m * 64 + k0];
      a.y  = w1[m * 64 + k0 + 1];
      f0.x = actB[k0][l15];        f0.y = actB[k0 + 1][l15];
      f1.x = actB[k0][16 + l15];   f1.y = actB[k0 + 1][16 + l15];
      c0 = wmma4(a, f0, c0);
      c1 = wmma4(a, f1, c1);
    }
#pragma unroll
    for (int v = 0; v < 8; ++v) {
      const int   r    = mt * 16 + v + 8 * half;
      const float scr  = g1[r] * rsqrtf(rv1[r] + EPS_);
      const float bias = scr * (b1[r] - rm1[r]) + bt1[r];
      actA[r][l15]      = fmaxf(fmaf(scr, c0[v], bias), 0.0f);
      actA[r][16 + l15] = fmaxf(fmaf(scr, c1[v], bias), 0.0f);
    }
  }
  __syncthreads();

  // ---- layer 2: 64 -> 128, read actA, fuse max over the 32 samples
  for (int mt = 0; mt < 8; ++mt) {
    const int m = mt * 16 + l15;
    v8f c0 = {}; v8f c1 = {};
#pragma unroll
    for (int ks = 0; ks < 16; ++ks) {
      const int k0 = ks * 4 + 2 * half;
      v2f a, f0, f1;
      a.x  = w2[m * 64 + k0];
      a.y  = w2[m * 64 + k0 + 1];
      f0.x = actA[k0][l15];        f0.y = actA[k0 + 1][l15];
      f1.x = actA[k0][16 + l15];   f1.y = actA[k0 + 1][16 + l15];
      c0 = wmma4(a, f0, c0);
      c1 = wmma4(a, f1, c1);
    }
#pragma unroll
    for (int v = 0; v < 8; ++v) {
      const int   r    = mt * 16 + v + 8 * half;
      const float scr  = g2[r] * rsqrtf(rv2[r] + EPS_);
      const float bias = scr * (b2[r] - rm2[r]) + bt2[r];
      float y = fmaxf(fmaxf(fmaf(scr, c0[v], bias), 0.0f),
                      fmaxf(fmaf(scr, c1[v], bias), 0.0f));
      // max across the 16 lanes of each half (columns of the D tile)
      y = fmaxf(y, __shfl_xor(y, 1, 32));
      y = fmaxf(y, __shfl_xor(y, 2, 32));
      y = fmaxf(y, __shfl_xor(y, 4, 32));
      y = fmaxf(y, __shfl_xor(y, 8, 32));
      if (l15 == 0)
        outF[((size_t)b * 128 + r) * S_ + s] = y;
    }
  }
}

// ---------------------------------------------------------------------------
extern "C" void kernel_launch(void* const* d_in, const int* in_sizes, int n_in,
                              void* d_out, int out_size, void* d_ws, size_t ws_size,
                              hipStream_t stream) {
  (void)in_sizes; (void)n_in; (void)out_size; (void)ws_size;
  const float* xyz    = (const float*)d_in[0];
  const float* points = (const float*)d_in[1];
  const float* w0  = (const float*)d_in[2];
  const float* b0  = (const float*)d_in[3];
  const float* g0  = (const float*)d_in[4];
  const float* bt0 = (const float*)d_in[5];
  const float* rm0 = (const float*)d_in[6];
  const float* rv0 = (const float*)d_in[7];
  const float* w1  = (const float*)d_in[8];
  const float* b1  = (const float*)d_in[9];
  const float* g1  = (const float*)d_in[10];
  const float* bt1 = (const float*)d_in[11];
  const float* rm1 = (const float*)d_in[12];
  const float* rv1 = (const float*)d_in[13];
  const float* w2  = (const float*)d_in[14];
  const float* b2  = (const float*)d_in[15];
  const float* g2  = (const float*)d_in[16];
  const float* bt2 = (const float*)d_in[17];
  const float* rm2 = (const float*)d_in[18];
  const float* rv2 = (const float*)d_in[19];

  float* out    = (float*)d_out;
  float* newxyz = out;                       // (B,3,S)
  float* feats  = out + (size_t)B_ * 3 * S_; // (B,128,S)
  int*   idx    = (int*)d_ws;                // (B,S,K_) = 1 MB scratch

  fps_kernel<<<B_, 256, 0, stream>>>(xyz, newxyz);
  ballquery_kernel<<<B_ * S_, 32, 0, stream>>>(xyz, newxyz, idx);
  fused_mlp_kernel<<<B_ * S_, 32, 0, stream>>>(
      xyz, points, idx, newxyz,
      w0, b0, g0, bt0, rm0, rv0,
      w1, b1, g1, bt1, rm1, rv1,
      w2, b2, g2, bt2, rm2, rv2,
      feats);
}